// BottomUp_6880537608269
// MI455X (gfx1250) — compile-verified
//
#include <hip/hip_runtime.h>
#include <stdint.h>

// ---------------------------------------------------------------------------
// Adding-doubling radiative transfer, B=2048, L=61, C=256, fp32.
// Memory-bound: 512MB in + 384MB out => ~38us floor at 23.3 TB/s.
// r,t staged once into LDS via the CDNA5 Tensor Data Mover (used by the two
// opposing-direction passes); a,s streamed non-temporally; every output
// element written exactly once. Scan state (rs/s_up, tru, amu, abs_down)
// lives in per-thread register arrays — round-2/3 histograms proved the
// allocator handles this without scratch spills, and volatile/barrier
// attempts to force LDS residency only added waits or spills.
// ---------------------------------------------------------------------------

typedef unsigned int u32;
typedef u32 v4u __attribute__((ext_vector_type(4)));
typedef int  v8i __attribute__((ext_vector_type(8)));
typedef int  v4i __attribute__((ext_vector_type(4)));

constexpr int L  = 61;
constexpr int C  = 256;
constexpr int CH = 128;          // channels handled per block (half of C)
constexpr int LC = L * C;

// Issue one TDM 2D tile load: rows x cols fp32 elements, global row stride
// 'stride' elements, packed contiguously into LDS at byte offset lds_off.
__device__ __forceinline__ void tdm_load_2d(u32 lds_off, const float* gptr,
                                            int rows, int cols, int stride) {
  uint64_t ga = (uint64_t)(uintptr_t)gptr;
  v4u g0;
  g0.x = 1u;                                        // count=1, user descriptor
  g0.y = lds_off;                                   // LDS byte address
  g0.z = (u32)(ga & 0xFFFFFFFFu);                   // global_addr[31:0]
  g0.w = (u32)((ga >> 32) & 0x01FFFFFFu)            // global_addr[56:32]
         | 0x80000000u;                             // type=2 ("image")
  v8i g1;
  g1[0] = (int)0x00020000u;                         // data_size=2 (4B)
  g1[1] = (int)(((u32)cols & 0xFFFFu) << 16);       // tensor_dim0[15:0]
  g1[2] = (int)((((u32)cols >> 16) & 0xFFFFu)       // tensor_dim0[31:16]
         | (((u32)rows & 0xFFFFu) << 16));          // tensor_dim1[15:0]
  g1[3] = (int)((((u32)rows >> 16) & 0xFFFFu)       // tensor_dim1[31:16]
         | (((u32)cols & 0xFFFFu) << 16));          // tile_dim0
  g1[4] = (int)((u32)rows & 0xFFFFu);               // tile_dim1 (tile_dim2=0)
  g1[5] = (int)(u32)stride;                         // tensor_dim0_stride[31:0]
  g1[6] = 0;                                        // stride hi / dim1_stride lo
  g1[7] = 0;
  v4i z4 = {0, 0, 0, 0};
  v8i z8 = {0, 0, 0, 0, 0, 0, 0, 0};
  // 6-arg variant (clang-23 / therock-10.0): (v4u, v8i, v4i, v4i, v8i, i32)
  __builtin_amdgcn_tensor_load_to_lds(g0, g1, z4, z4, z8, 0);
}

__global__ void __launch_bounds__(CH)
adding_doubling_kernel(const float* __restrict__ a, const float* __restrict__ r,
                       const float* __restrict__ t, const float* __restrict__ s,
                       float* __restrict__ out, int B) {
  __shared__ float lds_r[L * CH];   // r[l][c']
  __shared__ float lds_t[L * CH];   // t[l][c']

  const int bx   = blockIdx.x;
  const int b    = bx >> 1;
  const int c0   = (bx & 1) * CH;
  const int tid  = threadIdx.x;

  const size_t blkbase = (size_t)b * LC + c0;

  // ---- stage r,t tiles into LDS with the Tensor Data Mover (wave 0 only) ---
  if (tid < 32) {
    tdm_load_2d((u32)(uintptr_t)(void*)lds_r, r + blkbase, L, CH, C);
    tdm_load_2d((u32)(uintptr_t)(void*)lds_t, t + blkbase, L, CH, C);
    __builtin_amdgcn_s_wait_tensorcnt(0);
  }
  __syncthreads();

  const size_t colbase = blkbase + tid;
  const float* acol = a + colbase;
  const float* scol = s + colbase;
  const size_t N    = (size_t)B * LC;
  float* fd_out = out + colbase;          // flux_down
  float* fu_out = out + N + colbase;      // flux_up
  float* ab_out = out + 2 * N + colbase;  // absorbed

  // ---- Pass A (bottom-up): cumulative surface reflection rs[l] ------------
  float arr[L];                           // rs[l], later overwritten by s_up[l]
  arr[L - 1] = lds_r[(L - 1) * CH + tid];
#pragma unroll
  for (int l = L - 2; l >= 0; --l) {
    const float rl = lds_r[l * CH + tid];
    const float tl = lds_t[l * CH + tid];
    const float dd = __builtin_amdgcn_rcpf(1.0f - arr[l + 1] * rl);
    arr[l] = rl + arr[l + 1] * tl * tl * dd;
  }

  // ---- Pass B (top-down): rt/dt, downward flux + absorption, s_up/tru/amu -
  float tru[L - 1];
  float amu[L - 1];
  float absd[L];                          // abs_down[l]
  float rt_prev, s_prev, carry_d;
  {
    const float r0 = lds_r[tid];
    const float t0 = lds_t[tid];
    const float a0 = __builtin_nontemporal_load(acol);
    const float s0 = __builtin_nontemporal_load(scol);
    rt_prev = r0;
    s_prev  = s0;
    tru[0]  = t0;                          // t[0]*dt[0], dt[0]==1
    amu[0]  = a0;                          // a_mult_up[0]
    arr[0]  = s0;                          // s_up[0] = s[0]*d[0], d[0]==1
    absd[0] = 0.0f;                        // abs_down[0]
    __builtin_nontemporal_store(0.0f, fd_out);   // flux_down[0] = 0
    carry_d = 0.0f;
  }
#pragma unroll
  for (int l = 1; l < L; ++l) {
    const float rl = lds_r[l * CH + tid];
    const float tl = lds_t[l * CH + tid];
    const float al = __builtin_nontemporal_load(acol + (size_t)l * C);
    const float sl = __builtin_nontemporal_load(scol + (size_t)l * C);

    const float dt   = __builtin_amdgcn_rcpf(1.0f - rt_prev * rl);
    const float rs_l = arr[l];
    float ds_l, amd;
    if (l < L - 1) {
      const float rs_n = arr[l + 1];
      ds_l = __builtin_amdgcn_rcpf(1.0f - rs_n * rl);
      amd  = al * (1.0f + rs_n * tl * ds_l);       // a_mult_down
    } else {
      ds_l = 1.0f;
      amd  = al;                                   // a_mult_down[L-2] = a[L-1]
    }
    const float d_l = __builtin_amdgcn_rcpf(1.0f - rt_prev * rs_l);

    const float sdown = (s_prev + sl * rt_prev) * d_l;   // s_down[l-1]
    const float f_d   = carry_d + sdown;                  // flux_down[l]
    __builtin_nontemporal_store(f_d, fd_out + (size_t)l * C);
    absd[l] = amd * f_d;                                  // abs_down[l]
    carry_d = f_d * tl * ds_l;

    arr[l] = (sl + s_prev * rs_l) * d_l;                  // s_up[l]
    if (l < L - 1) {
      tru[l] = tl * dt;                                   // t[l]*dt[l]
      amu[l] = al * (1.0f + tl * rt_prev * dt);           // a_mult_up[l]
    }
    rt_prev = rl + rt_prev * tl * tl * dt;                // rt[l]
    s_prev  = sl;
  }

  // ---- Pass C (bottom-up): upward flux; absorbed = abs_down + abs_up ------
  float carry_u = 0.0f;
#pragma unroll
  for (int l = L - 1; l >= 1; --l) {
    const float f_u = carry_u + arr[l];                   // flux_up[l]
    __builtin_nontemporal_store(f_u, fu_out + (size_t)l * C);
    const float absv = absd[l - 1] + amu[l - 1] * f_u;    // + abs_up[l-1]
    __builtin_nontemporal_store(absv, ab_out + (size_t)(l - 1) * C);
    carry_u = f_u * tru[l - 1];
  }
  __builtin_nontemporal_store(carry_u + arr[0], fu_out);  // flux_up[0]
  __builtin_nontemporal_store(absd[L - 1],                // abs_up[L-1]==0
                              ab_out + (size_t)(L - 1) * C);
}

extern "C" void kernel_launch(void* const* d_in, const int* in_sizes, int n_in,
                              void* d_out, int out_size, void* d_ws, size_t ws_size,
                              hipStream_t stream) {
  const float* a = (const float*)d_in[0];
  const float* r = (const float*)d_in[1];
  const float* t = (const float*)d_in[2];
  const float* s = (const float*)d_in[3];
  float* out = (float*)d_out;
  const int B = in_sizes[0] / (L * C);            // 2048
  dim3 grid((unsigned)(B * (C / CH)));            // 4096 blocks
  dim3 block(CH);                                 // 128 threads (4 waves)
  adding_doubling_kernel<<<grid, block, 0, stream>>>(a, r, t, s, out, B);
}